// VascularGCN_34127810134070
// MI455X (gfx1250) — compile-verified
//
#include <hip/hip_runtime.h>
#include <hip/hip_bf16.h>

// ---------------------------------------------------------------------------
// Sizes (fixed by the reference)
// ---------------------------------------------------------------------------
#define N_NODES  100000
#define N_EDGES  1600000
#define N_GRAPHS 64
#define IN_C     12
#define HID      64
#define BN_EPS   1e-5f

typedef __attribute__((ext_vector_type(2))) float v2f;
typedef __attribute__((ext_vector_type(8))) float v8f;

// ---------------------------------------------------------------------------
// Utility kernels
// ---------------------------------------------------------------------------
__global__ void fill_kernel(float* __restrict__ p, long n, float v) {
    long t = (long)blockIdx.x * blockDim.x + threadIdx.x;
    if (t < n) p[t] = v;
}

// vectorized fill: n4 float4 elements (use only on 16B-aligned, n%4==0 buffers)
__global__ void fill4_kernel(float4* __restrict__ p, long n4, float v) {
    long t = (long)blockIdx.x * blockDim.x + threadIdx.x;
    if (t < n4) p[t] = make_float4(v, v, v, v);
}

__global__ void deg_kernel(const int* __restrict__ ei, float* __restrict__ deg, int E) {
    long t = (long)blockIdx.x * blockDim.x + threadIdx.x;
    if (t >= E) return;
    int d = ei[E + t];               // dst row of edge_index
    atomicAdd(&deg[d], 1.0f);
}

__global__ void rsqrt_kernel(float* __restrict__ d, int n) {
    long t = (long)blockIdx.x * blockDim.x + threadIdx.x;
    if (t < n) d[t] = rsqrtf(d[t]);
}

__global__ void enorm_kernel(const int* __restrict__ ei, const float* __restrict__ dis,
                             float* __restrict__ enorm, int E) {
    long t = (long)blockIdx.x * blockDim.x + threadIdx.x;
    if (t >= E) return;
    int s = ei[t];
    int d = ei[E + t];
    enorm[t] = dis[s] * dis[d];
}

// ---------------------------------------------------------------------------
// WMMA f32 GEMM:  C[N,64] = A[N,K] @ W[K,64]   using V_WMMA_F32_16X16X4_F32
// One wave computes a 16-row x 64-col tile (4 accumulators of 16 cols each).
// A-fragment layout (ISA 7.12.2, 32-bit A 16x4): lanes 0-15 hold K=k0,k0+1,
// lanes 16-31 hold K=k0+2,k0+3 for rows M=lane%16.
// B-fragment (4x16, KxN): lanes 0-15 rows K=k0,(VGPR1)k0+1 ; lanes 16-31
// rows K=k0+2,k0+3 ; N = lane%16.
// C/D: VGPR v -> M=v (lanes 0-15) / M=v+8 (lanes 16-31), N=lane%16.
// ---------------------------------------------------------------------------
template <int K>
__global__ void gemm_wmma_kernel(const float* __restrict__ A,
                                 const float* __restrict__ W,
                                 float* __restrict__ C,
                                 int nRows) {
    const int lane   = threadIdx.x & 31;
    const int wave   = (blockIdx.x * (blockDim.x >> 5)) + (threadIdx.x >> 5);
    const int rowBase = wave * 16;
    if (rowBase >= nRows) return;          // wave-uniform: EXEC stays all-1 in live waves

    const int half = lane >> 4;            // 0: lanes 0-15, 1: lanes 16-31
    const int l15  = lane & 15;
    const int arow = rowBase + l15;

    v8f acc0 = {0,0,0,0,0,0,0,0};
    v8f acc1 = {0,0,0,0,0,0,0,0};
    v8f acc2 = {0,0,0,0,0,0,0,0};
    v8f acc3 = {0,0,0,0,0,0,0,0};

#pragma unroll
    for (int k0 = 0; k0 < K; k0 += 4) {
        const int ka = k0 + 2 * half;
        v2f a;
        a.x = A[(long)arow * K + ka];
        a.y = A[(long)arow * K + ka + 1];

        v2f b0, b1, b2, b3;
        b0.x = W[ka * HID +  0 + l15];  b0.y = W[(ka + 1) * HID +  0 + l15];
        b1.x = W[ka * HID + 16 + l15];  b1.y = W[(ka + 1) * HID + 16 + l15];
        b2.x = W[ka * HID + 32 + l15];  b2.y = W[(ka + 1) * HID + 32 + l15];
        b3.x = W[ka * HID + 48 + l15];  b3.y = W[(ka + 1) * HID + 48 + l15];

        acc0 = __builtin_amdgcn_wmma_f32_16x16x4_f32(false, a, false, b0, (short)0, acc0, false, false);
        acc1 = __builtin_amdgcn_wmma_f32_16x16x4_f32(false, a, false, b1, (short)0, acc1, false, false);
        acc2 = __builtin_amdgcn_wmma_f32_16x16x4_f32(false, a, false, b2, (short)0, acc2, false, false);
        acc3 = __builtin_amdgcn_wmma_f32_16x16x4_f32(false, a, false, b3, (short)0, acc3, false, false);
    }

#pragma unroll
    for (int v = 0; v < 8; ++v) {
        const int row = rowBase + v + half * 8;
        float* out = C + (long)row * HID + l15;
        out[ 0] = acc0[v];
        out[16] = acc1[v];
        out[32] = acc2[v];
        out[48] = acc3[v];
    }
}

// ---------------------------------------------------------------------------
// Edge scatter:  agg[dst, c] += hw[src, c] * enorm[e]
// One thread per (edge, channel-quad): 1x global_load_b128 + 4 f32 atomics.
// hw rows are 64 floats (256 B) so every 4-channel quad is 16B-aligned.
// ---------------------------------------------------------------------------
__global__ void scatter_kernel(const int* __restrict__ ei,
                               const float* __restrict__ hw,
                               const float* __restrict__ enorm,
                               float* __restrict__ agg, int E) {
    long t = (long)blockIdx.x * blockDim.x + threadIdx.x;
    if (t >= (long)E * (HID / 4)) return;
    int e = (int)(t >> 4);            // 16 quads per edge
    int q = (int)(t & 15);
    int s = ei[e];
    int d = ei[E + e];
    float w = enorm[e];
    const float4 hv = *(const float4*)(hw + (long)s * HID + q * 4);
    float* ap = agg + (long)d * HID + q * 4;
    atomicAdd(ap + 0, hv.x * w);
    atomicAdd(ap + 1, hv.y * w);
    atomicAdd(ap + 2, hv.z * w);
    atomicAdd(ap + 3, hv.w * w);
}

// ---------------------------------------------------------------------------
// h_next = relu( ((agg + hw*self_norm + b) - rm) * rsqrt(rv+eps) * g + bt )
// Vectorized over channel quads; safe in-place (out may alias agg).
// ---------------------------------------------------------------------------
__global__ void update_kernel(const float4* __restrict__ agg,
                              const float4* __restrict__ hw,
                              const float* __restrict__ dis,
                              const float4* __restrict__ b,
                              const float4* __restrict__ g,
                              const float4* __restrict__ bt,
                              const float4* __restrict__ rm,
                              const float4* __restrict__ rv,
                              float4* __restrict__ out, int n) {
    long t = (long)blockIdx.x * blockDim.x + threadIdx.x;
    if (t >= (long)n * (HID / 4)) return;
    int i = (int)(t >> 4);
    int q = (int)(t & 15);
    float sn = dis[i] * dis[i];               // 1/deg self-loop norm
    float4 av = agg[t];
    float4 hv = hw[t];
    float4 bv = b[q],  gv = g[q],  btv = bt[q], rmv = rm[q], rvv = rv[q];
    float4 r;
    r.x = fmaxf((av.x + hv.x * sn + bv.x - rmv.x) * rsqrtf(rvv.x + BN_EPS) * gv.x + btv.x, 0.0f);
    r.y = fmaxf((av.y + hv.y * sn + bv.y - rmv.y) * rsqrtf(rvv.y + BN_EPS) * gv.y + btv.y, 0.0f);
    r.z = fmaxf((av.z + hv.z * sn + bv.z - rmv.z) * rsqrtf(rvv.z + BN_EPS) * gv.z + btv.z, 0.0f);
    r.w = fmaxf((av.w + hv.w * sn + bv.w - rmv.w) * rsqrtf(rvv.w + BN_EPS) * gv.w + btv.w, 0.0f);
    out[t] = r;
}

// ---------------------------------------------------------------------------
// Global mean pool accumulation (channel quads, b128 load + 4 atomics)
// ---------------------------------------------------------------------------
__global__ void pool_kernel(const float4* __restrict__ h,
                            const int* __restrict__ batch,
                            float* __restrict__ sums,
                            float* __restrict__ cnts, int n) {
    long t = (long)blockIdx.x * blockDim.x + threadIdx.x;
    if (t >= (long)n * (HID / 4)) return;
    int i = (int)(t >> 4);
    int q = (int)(t & 15);
    int gidx = batch[i];
    float4 hv = h[t];
    float* sp = sums + (long)gidx * HID + q * 4;
    atomicAdd(sp + 0, hv.x);
    atomicAdd(sp + 1, hv.y);
    atomicAdd(sp + 2, hv.z);
    atomicAdd(sp + 3, hv.w);
    if (q == 0) atomicAdd(&cnts[gidx], 1.0f);
}

// ---------------------------------------------------------------------------
// Heads: graph_emb -> classifier (64->32->2) and embedding (64->64)
// Tiny: one thread per graph.
// ---------------------------------------------------------------------------
__global__ void head_kernel(const float* __restrict__ sums,
                            const float* __restrict__ cnts,
                            const float* __restrict__ Wc1, const float* __restrict__ bc1,
                            const float* __restrict__ Wc2, const float* __restrict__ bc2,
                            const float* __restrict__ We,  const float* __restrict__ be,
                            float* __restrict__ logits,     // [64,2]
                            float* __restrict__ embedding)  // [64,64]
{
    int g = threadIdx.x;
    if (g >= N_GRAPHS) return;
    float cnt = fmaxf(cnts[g], 1.0f);
    float ge[HID];
#pragma unroll
    for (int c = 0; c < HID; ++c) ge[c] = sums[g * HID + c] / cnt;

    float hid[HID / 2];
    for (int j = 0; j < HID / 2; ++j) {
        float s = bc1[j];
        for (int c = 0; c < HID; ++c) s += ge[c] * Wc1[c * (HID / 2) + j];
        hid[j] = fmaxf(s, 0.0f);
    }
    for (int j = 0; j < 2; ++j) {
        float s = bc2[j];
        for (int c = 0; c < HID / 2; ++c) s += hid[c] * Wc2[c * 2 + j];
        logits[g * 2 + j] = s;
    }
    for (int j = 0; j < HID; ++j) {
        float s = be[j];
        for (int c = 0; c < HID; ++c) s += ge[c] * We[c * HID + j];
        embedding[g * HID + j] = s;
    }
}

// ---------------------------------------------------------------------------
// Launch
// ---------------------------------------------------------------------------
extern "C" void kernel_launch(void* const* d_in, const int* in_sizes, int n_in,
                              void* d_out, int out_size, void* d_ws, size_t ws_size,
                              hipStream_t stream) {
    (void)in_sizes; (void)n_in; (void)out_size; (void)ws_size;

    const float* x     = (const float*)d_in[0];
    const int*   ei    = (const int*)d_in[1];
    const int*   batch = (const int*)d_in[2];
    const float* W[3]  = {(const float*)d_in[3],  (const float*)d_in[9],  (const float*)d_in[15]};
    const float* b[3]  = {(const float*)d_in[4],  (const float*)d_in[10], (const float*)d_in[16]};
    const float* gam[3]= {(const float*)d_in[5],  (const float*)d_in[11], (const float*)d_in[17]};
    const float* bt[3] = {(const float*)d_in[6],  (const float*)d_in[12], (const float*)d_in[18]};
    const float* rm[3] = {(const float*)d_in[7],  (const float*)d_in[13], (const float*)d_in[19]};
    const float* rv[3] = {(const float*)d_in[8],  (const float*)d_in[14], (const float*)d_in[20]};
    const float* Wc1 = (const float*)d_in[21];
    const float* bc1 = (const float*)d_in[22];
    const float* Wc2 = (const float*)d_in[23];
    const float* bc2 = (const float*)d_in[24];
    const float* We  = (const float*)d_in[25];
    const float* be  = (const float*)d_in[26];

    float* out       = (float*)d_out;
    float* logits    = out;                        // [64,2]
    float* embedding = out + N_GRAPHS * 2;         // [64,64]
    float* nodeEmb   = embedding + N_GRAPHS * HID; // [100000,64]

    // workspace layout (floats; all big buffers 16B-aligned by construction)
    float* ws = (float*)d_ws;
    size_t off = 0;
    float* dis   = ws + off; off += 100352;                 // deg -> dis (padded)
    float* enorm = ws + off; off += (size_t)N_EDGES;        // per-edge norm
    float* bufA  = ws + off; off += (size_t)N_NODES * HID;  // hw
    float* bufB  = ws + off; off += (size_t)N_NODES * HID;  // agg / h ping
    float* bufC  = ws + off; off += (size_t)N_NODES * HID;  // agg / h pong
    float* sums  = ws + off; off += (size_t)N_GRAPHS * HID;
    float* cnts  = ws + off; off += (size_t)N_GRAPHS;

    const int TPB = 256;
    const long NH4 = (long)N_NODES * (HID / 4);   // 1,600,000 quads
    const long EH4 = (long)N_EDGES * (HID / 4);   // 25,600,000 quads
    const long NHv = (long)N_NODES * HID / 4;     // float4 count for fills
    const int gN    = (int)((N_NODES + TPB - 1) / TPB);
    const int gE    = (int)((N_EDGES + TPB - 1) / TPB);
    const int gNH4  = (int)((NH4 + TPB - 1) / TPB);
    const int gEH4  = (int)((EH4 + TPB - 1) / TPB);
    const int gFill = (int)((NHv + TPB - 1) / TPB);
    // GEMM: 1 wave = 16 rows, 4 waves per block of 128 threads
    const int waves  = N_NODES / 16;                    // 6250 (exact)
    const int gGemm  = (waves + 3) / 4;                 // 1563 blocks

    // 1) degree (with self loop) -> dis -> enorm
    fill_kernel<<<gN, TPB, 0, stream>>>(dis, N_NODES, 1.0f);
    deg_kernel<<<gE, TPB, 0, stream>>>(ei, dis, N_EDGES);
    rsqrt_kernel<<<gN, TPB, 0, stream>>>(dis, N_NODES);
    enorm_kernel<<<gE, TPB, 0, stream>>>(ei, dis, enorm, N_EDGES);

    // 2) layer 0: x[N,12] @ W0 -> bufA ; scatter -> bufB ; update in-place bufB
    gemm_wmma_kernel<IN_C><<<gGemm, 128, 0, stream>>>(x, W[0], bufA, N_NODES);
    fill4_kernel<<<gFill, TPB, 0, stream>>>((float4*)bufB, NHv, 0.0f);
    scatter_kernel<<<gEH4, TPB, 0, stream>>>(ei, bufA, enorm, bufB, N_EDGES);
    update_kernel<<<gNH4, TPB, 0, stream>>>((const float4*)bufB, (const float4*)bufA, dis,
                                            (const float4*)b[0], (const float4*)gam[0],
                                            (const float4*)bt[0], (const float4*)rm[0],
                                            (const float4*)rv[0], (float4*)bufB, N_NODES);

    // 3) layer 1: bufB @ W1 -> bufA ; scatter -> bufC ; update in-place bufC
    gemm_wmma_kernel<HID><<<gGemm, 128, 0, stream>>>(bufB, W[1], bufA, N_NODES);
    fill4_kernel<<<gFill, TPB, 0, stream>>>((float4*)bufC, NHv, 0.0f);
    scatter_kernel<<<gEH4, TPB, 0, stream>>>(ei, bufA, enorm, bufC, N_EDGES);
    update_kernel<<<gNH4, TPB, 0, stream>>>((const float4*)bufC, (const float4*)bufA, dis,
                                            (const float4*)b[1], (const float4*)gam[1],
                                            (const float4*)bt[1], (const float4*)rm[1],
                                            (const float4*)rv[1], (float4*)bufC, N_NODES);

    // 4) layer 2: bufC @ W2 -> bufA ; scatter -> bufB ; update -> node embeddings (d_out)
    gemm_wmma_kernel<HID><<<gGemm, 128, 0, stream>>>(bufC, W[2], bufA, N_NODES);
    fill4_kernel<<<gFill, TPB, 0, stream>>>((float4*)bufB, NHv, 0.0f);
    scatter_kernel<<<gEH4, TPB, 0, stream>>>(ei, bufA, enorm, bufB, N_EDGES);
    update_kernel<<<gNH4, TPB, 0, stream>>>((const float4*)bufB, (const float4*)bufA, dis,
                                            (const float4*)b[2], (const float4*)gam[2],
                                            (const float4*)bt[2], (const float4*)rm[2],
                                            (const float4*)rv[2], (float4*)nodeEmb, N_NODES);

    // 5) pool + heads  (sums and cnts are contiguous -> one fill)
    fill_kernel<<<(N_GRAPHS * HID + N_GRAPHS + TPB - 1) / TPB, TPB, 0, stream>>>(
        sums, N_GRAPHS * HID + N_GRAPHS, 0.0f);
    pool_kernel<<<gNH4, TPB, 0, stream>>>((const float4*)nodeEmb, batch, sums, cnts, N_NODES);
    head_kernel<<<1, 64, 0, stream>>>(sums, cnts, Wc1, bc1, Wc2, bc2, We, be,
                                      logits, embedding);
}